// GCNII_29712583754280
// MI455X (gfx1250) — compile-verified
//
#include <hip/hip_runtime.h>
#include <math.h>

#define HID     128
#define IN_DIM  256
#define NLAYERS 8
#define ALPHA_C 0.1f
#define THETA_C 0.5f

typedef __attribute__((ext_vector_type(2))) float v2f;
typedef __attribute__((ext_vector_type(8))) float v8f;

// ---------------- CSR build ----------------

__global__ void k_zero_i32(int* __restrict__ p, int n) {
    int i = blockIdx.x * blockDim.x + threadIdx.x;
    if (i < n) p[i] = 0;
}

// counts[dst]++ per edge (int atomics: exact & deterministic)
__global__ void k_hist(const long long* __restrict__ dst, int* __restrict__ cnt, int E) {
    int e = blockIdx.x * blockDim.x + threadIdx.x;
    if (e < E) atomicAdd(&cnt[(int)dst[e]], 1);
}

// per-256-block exclusive scan; block total to bsum
__global__ void k_scan_block(const int* __restrict__ cnt, int* __restrict__ out,
                             int* __restrict__ bsum, int n) {
    __shared__ int sm[256];
    int i = blockIdx.x * 256 + threadIdx.x;
    int v = (i < n) ? cnt[i] : 0;
    sm[threadIdx.x] = v;
    __syncthreads();
    for (int off = 1; off < 256; off <<= 1) {
        int t = (threadIdx.x >= off) ? sm[threadIdx.x - off] : 0;
        __syncthreads();
        sm[threadIdx.x] += t;
        __syncthreads();
    }
    int incl = sm[threadIdx.x];
    if (i < n) out[i] = incl - v;                 // exclusive
    if (threadIdx.x == 255) bsum[blockIdx.x] = incl;
}

// single-block exclusive scan of the block sums (nb <= 512)
__global__ void k_scan_single(int* __restrict__ bsum, int nb) {
    __shared__ int sm[512];
    int v = (threadIdx.x < nb) ? bsum[threadIdx.x] : 0;
    sm[threadIdx.x] = v;
    __syncthreads();
    for (int off = 1; off < 512; off <<= 1) {
        int t = (threadIdx.x >= (unsigned)off) ? sm[threadIdx.x - off] : 0;
        __syncthreads();
        sm[threadIdx.x] += t;
        __syncthreads();
    }
    if (threadIdx.x < nb) bsum[threadIdx.x] = sm[threadIdx.x] - v;
}

// rowptr += scanned block offset; cursor = rowptr copy; rowptr[n] = E
__global__ void k_scan_add(int* __restrict__ rowptr, const int* __restrict__ bsum,
                           int* __restrict__ cursor, int n, int E) {
    int i = blockIdx.x * blockDim.x + threadIdx.x;
    if (i < n) {
        int r = rowptr[i] + bsum[i >> 8];
        rowptr[i] = r;
        cursor[i] = r;
    }
    if (i == 0) rowptr[n] = E;
}

// dinv[i] = rsqrt(deg_in + 1)  (self loop folded in)
__global__ void k_dinv(const int* __restrict__ rowptr, float* __restrict__ dinv, int n) {
    int i = blockIdx.x * blockDim.x + threadIdx.x;
    if (i < n) dinv[i] = rsqrtf((float)(rowptr[i + 1] - rowptr[i]) + 1.0f);
}

// col[slot] = src for each edge (slot via atomic cursor; order fixed later by sort)
__global__ void k_fill(const long long* __restrict__ ei, int* __restrict__ cursor,
                       int* __restrict__ col, int E) {
    int e = blockIdx.x * blockDim.x + threadIdx.x;
    if (e < E) {
        int s = (int)ei[e];
        int d = (int)ei[(size_t)E + e];
        int p = atomicAdd(&cursor[d], 1);
        col[p] = s;
    }
}

// one thread per row: insertion-sort cols (avg degree ~10) for bit-determinism,
// then precompute layer-invariant edge weights wgt = dinv[src]*dinv[dst]
__global__ void k_rowfix(const int* __restrict__ rowptr, int* __restrict__ col,
                         float* __restrict__ wgt, const float* __restrict__ dinv, int n) {
    int i = blockIdx.x * blockDim.x + threadIdx.x;
    if (i >= n) return;
    int b = rowptr[i], e = rowptr[i + 1];
    for (int p = b + 1; p < e; ++p) {
        int key = col[p];
        int q = p - 1;
        while (q >= b && col[q] > key) { col[q + 1] = col[q]; --q; }
        col[q + 1] = key;
    }
    float di = dinv[i];
    for (int p = b; p < e; ++p) wgt[p] = di * dinv[col[p]];
}

// ---------------- fused aggregation (atomic-free gather) ----------------
// one wave per dst node; lane owns a float4 chunk of the 128-dim feature.
// comb[i] = (1-alpha) * ( h[i]*dinv[i]^2 + sum_e wgt[e]*h[col[e]] ) + alpha*h0[i]
__global__ void k_aggregate(const float* __restrict__ h, const float* __restrict__ h0,
                            const float* __restrict__ dinv, const int* __restrict__ rowptr,
                            const int* __restrict__ col, const float* __restrict__ wgt,
                            float* __restrict__ comb, int n) {
    int gw   = (blockIdx.x * blockDim.x + threadIdx.x) >> 5;   // global wave = node
    int lane = threadIdx.x & 31;
    if (gw >= n) return;
    const float4* __restrict__ h4 = (const float4*)h;

    float di = dinv[gw];
    float w2 = di * di;
    float4 acc = h4[(size_t)gw * 32 + lane];
    acc.x *= w2; acc.y *= w2; acc.z *= w2; acc.w *= w2;

    int b = rowptr[gw], e = rowptr[gw + 1];
    for (int p = b; p < e; ++p) {
        int   s = col[p];            // broadcast load (uniform across wave)
        float w = wgt[p];
        float4 v = h4[(size_t)s * 32 + lane];
        acc.x += w * v.x; acc.y += w * v.y; acc.z += w * v.z; acc.w += w * v.w;
    }

    float4 z = ((const float4*)h0)[(size_t)gw * 32 + lane];
    float4 r;
    r.x = (1.0f - ALPHA_C) * acc.x + ALPHA_C * z.x;
    r.y = (1.0f - ALPHA_C) * acc.y + ALPHA_C * z.y;
    r.z = (1.0f - ALPHA_C) * acc.z + ALPHA_C * z.z;
    r.w = (1.0f - ALPHA_C) * acc.w + ALPHA_C * z.w;
    ((float4*)comb)[(size_t)gw * 32 + lane] = r;
}

// ---------------- fp32 WMMA GEMM ----------------
// D[r][c] = epilogue( s0 * (A @ W)[r][c] + s1 * A[r][c] + bias[c] )
// One wave computes a 16x16 tile via V_WMMA_F32_16X16X4_F32; 8 waves/block
// cover all HID=128 output columns of a 16-row stripe.
template <int K>
__global__ void k_gemm(const float* __restrict__ A, const float* __restrict__ W,
                       const float* __restrict__ bias, float* __restrict__ D,
                       float* __restrict__ D2, int M, float s0, float s1, int do_relu) {
    const int lane = threadIdx.x & 31;
    const int wave = threadIdx.x >> 5;
    const int row0 = blockIdx.x * 16;
    if (row0 >= M) return;                    // uniform per block: EXEC stays all-ones
    const int col0 = wave * 16;
    const int m    = lane & 15;               // A: lane holds row m
    const int nn   = lane & 15;               // B/D: lane holds col nn
    const int half = lane >> 4;
    const int kb   = half << 1;               // K phase within the 16x16x4 fragment

    const float* __restrict__ arow = A + (size_t)(row0 + m) * K + kb;
    const float* __restrict__ bcol = W + (size_t)kb * HID + col0 + nn;

    v8f acc = {};
#pragma unroll 4
    for (int k = 0; k < K; k += 4) {
        v2f a; a.x = arow[0];  a.y = arow[1];     // A[m][k+kb], A[m][k+kb+1]
        v2f b; b.x = bcol[0];  b.y = bcol[HID];   // W[k+kb][c], W[k+kb+1][c]
        acc = __builtin_amdgcn_wmma_f32_16x16x4_f32(
            /*neg_a=*/false, a, /*neg_b=*/false, b,
            /*c_mod=*/(short)0, acc, /*reuse_a=*/false, /*reuse_b=*/false);
        arow += 4;
        bcol += 4 * HID;
    }

    const int c  = col0 + nn;
    const float bb = bias ? bias[c] : 0.0f;
#pragma unroll
    for (int i = 0; i < 8; ++i) {
        int r = row0 + i + half * 8;             // C/D layout: VGPR i -> rows i / i+8
        float v = s0 * acc[i] + bb;
        if (s1 != 0.0f) v += s1 * A[(size_t)r * K + c];   // only used when K==HID
        if (do_relu) v = fmaxf(v, 0.0f);
        D[(size_t)r * HID + c] = v;
        if (D2) D2[(size_t)r * HID + c] = v;
    }
}

// ---------------- host orchestration ----------------

extern "C" void kernel_launch(void* const* d_in, const int* in_sizes, int n_in,
                              void* d_out, int out_size, void* d_ws, size_t ws_size,
                              hipStream_t stream) {
    const float*     x      = (const float*)d_in[0];
    const long long* ei     = (const long long*)d_in[1];
    const float*     W1     = (const float*)d_in[2];
    const float*     b1     = (const float*)d_in[3];
    const float*     conv_w = (const float*)d_in[4];
    const float*     W2     = (const float*)d_in[5];
    const float*     b2     = (const float*)d_in[6];
    const float*     Wv     = (const float*)d_in[7];
    const float*     bvp    = (const float*)d_in[8];
    const float*     Wt     = (const float*)d_in[9];
    const float*     btp    = (const float*)d_in[10];

    const int n = in_sizes[0] / IN_DIM;       // 100000
    const int E = in_sizes[1] / 2;            // 1000000

    // workspace layout (16B-aligned slabs)
    char* ws = (char*)d_ws;
    size_t off = 0;
    auto alloc = [&](size_t bytes) { void* p = ws + off; off += (bytes + 255) & ~(size_t)255; return p; };
    int*   rowptr = (int*)  alloc(((size_t)n + 1) * 4);
    int*   cursor = (int*)  alloc((size_t)n * 4);
    int*   bsum   = (int*)  alloc(512 * 4);
    int*   col    = (int*)  alloc((size_t)E * 4);
    float* wgt    = (float*)alloc((size_t)E * 4);
    float* dinv   = (float*)alloc((size_t)n * 4);
    float* h      = (float*)alloc((size_t)n * HID * 4);
    float* h0     = (float*)alloc((size_t)n * HID * 4);
    float* agg    = (float*)alloc((size_t)n * HID * 4);

    float* out = (float*)d_out;
    float* xv  = out + (size_t)n * HID;
    float* xt  = xv  + (size_t)n * HID;

    const int B     = 256;
    const int gN    = (n + B - 1) / B;
    const int gE    = (E + B - 1) / B;
    const int nb    = (n + 255) / 256;                 // scan blocks (391 <= 512)
    const int gAgg  = ((n * 32) + B - 1) / B;          // wave per node
    const int gGemm = (n + 15) / 16;                   // 16-row stripes

    // --- CSR build + gcn_norm ---
    k_zero_i32<<<gN, B, 0, stream>>>(cursor, n);                 // cursor as counts
    k_hist<<<gE, B, 0, stream>>>(ei + E, cursor, E);             // dst = edge_index[1]
    k_scan_block<<<nb, 256, 0, stream>>>(cursor, rowptr, bsum, n);
    k_scan_single<<<1, 512, 0, stream>>>(bsum, nb);
    k_scan_add<<<gN, B, 0, stream>>>(rowptr, bsum, cursor, n, E);
    k_dinv<<<gN, B, 0, stream>>>(rowptr, dinv, n);
    k_fill<<<gE, B, 0, stream>>>(ei, cursor, col, E);
    k_rowfix<<<gN, B, 0, stream>>>(rowptr, col, wgt, dinv, n);

    // --- h = relu(x @ W1 + b1), also h0 = h ---
    k_gemm<IN_DIM><<<gGemm, B, 0, stream>>>(x, W1, b1, h, h0, n, 1.0f, 0.0f, 1);

    // --- 8 GCNII layers: fused gather-aggregate + WMMA GEMM ---
    for (int l = 0; l < NLAYERS; ++l) {
        float beta = logf(THETA_C / (float)(l + 1) + 1.0f);
        k_aggregate<<<gAgg, B, 0, stream>>>(h, h0, dinv, rowptr, col, wgt, agg, n);
        // h = relu(beta * (agg @ Wl) + (1-beta) * agg)
        k_gemm<HID><<<gGemm, B, 0, stream>>>(agg, conv_w + (size_t)l * HID * HID,
                                             nullptr, h, nullptr, n, beta, 1.0f - beta, 1);
    }

    // --- heads ---
    k_gemm<HID><<<gGemm, B, 0, stream>>>(h,   W2, b2,  out, nullptr, n, 1.0f, 0.0f, 0);
    k_gemm<HID><<<gGemm, B, 0, stream>>>(out, Wv, bvp, xv,  nullptr, n, 1.0f, 0.0f, 1);
    k_gemm<HID><<<gGemm, B, 0, stream>>>(out, Wt, btp, xt,  nullptr, n, 1.0f, 0.0f, 1);
}